// TemporalGNN_52578989638197
// MI455X (gfx1250) — compile-verified
//
#include <hip/hip_runtime.h>
#include <math.h>

typedef __attribute__((ext_vector_type(2))) float v2f;
typedef __attribute__((ext_vector_type(8))) float v8f;

// ---------------------------------------------------------------- utilities
__global__ void zero_f32_kernel(float* __restrict__ p, long long count) {
    long long i = (long long)blockIdx.x * blockDim.x + threadIdx.x;
    long long stride = (long long)gridDim.x * blockDim.x;
    for (; i < count; i += stride) p[i] = 0.0f;
}

// degree counts via float atomics (values are small integers, exact in f32)
__global__ void degree_kernel(const int* __restrict__ src, const int* __restrict__ dst,
                              float* __restrict__ deg_out, float* __restrict__ deg_in, int E) {
    int i = blockIdx.x * blockDim.x + threadIdx.x;
    int stride = gridDim.x * blockDim.x;
    for (; i < E; i += stride) {
        atomicAdd(&deg_out[src[i]], 1.0f);
        atomicAdd(&deg_in[dst[i]], 1.0f);
    }
}

// in-place deg -> rsqrt(max(deg,1)) over both arrays (2N contiguous floats)
__global__ void invsqrt_kernel(float* __restrict__ d, int count) {
    int i = blockIdx.x * blockDim.x + threadIdx.x;
    int stride = gridDim.x * blockDim.x;
    for (; i < count; i += stride) d[i] = rsqrtf(fmaxf(d[i], 1.0f));
}

// ---------------------------------------------------------------- conv0 (F=3)
// gather feat[src]*inv_out[src], scatter-add into agg0 (stride 4 padded)
__global__ void agg3_kernel(const float* __restrict__ feat, const int* __restrict__ src,
                            const int* __restrict__ dst, const float* __restrict__ inv_out,
                            float* __restrict__ agg0, int E) {
    int e = blockIdx.x * blockDim.x + threadIdx.x;
    int stride = gridDim.x * blockDim.x;
    for (; e < E; e += stride) {
        const int s = src[e], d = dst[e];
        const float sc = inv_out[s];
        const float* fp = feat + (size_t)s * 3;
        float* op = agg0 + (size_t)d * 4;
        atomicAdd(op + 0, fp[0] * sc);
        atomicAdd(op + 1, fp[1] * sc);
        atomicAdd(op + 2, fp[2] * sc);
    }
}

// hs = relu((agg0 * inv_in) @ W0 + b0) * inv_out   (conv0 output, pre-scaled for conv1)
__global__ void conv0_kernel(const float* __restrict__ agg0, const float* __restrict__ inv_in,
                             const float* __restrict__ inv_out, const float* __restrict__ W0,
                             const float* __restrict__ b0, float* __restrict__ hs, int N) {
    int idx = blockIdx.x * blockDim.x + threadIdx.x;
    const int total = N * 64;
    const int stride = gridDim.x * blockDim.x;
    for (; idx < total; idx += stride) {
        const int n = idx >> 6, f = idx & 63;
        const float ii = inv_in[n];
        const float a0 = agg0[(size_t)n * 4 + 0] * ii;
        const float a1 = agg0[(size_t)n * 4 + 1] * ii;
        const float a2 = agg0[(size_t)n * 4 + 2] * ii;
        float v = a0 * W0[f] + a1 * W0[64 + f] + a2 * W0[128 + f] + b0[f];
        v = v > 0.0f ? v : 0.0f;
        hs[idx] = v * inv_out[n];
    }
}

// ---------------------------------------------------------------- conv1 aggregation (F=64)
// one wave per edge, 2 floats per lane; L2-resident float atomics
__global__ void agg64_kernel(const float* __restrict__ hs, const int* __restrict__ src,
                             const int* __restrict__ dst, float* __restrict__ agg1, int E) {
    const int lane = threadIdx.x & 31;
    int wave = blockIdx.x * (blockDim.x >> 5) + (threadIdx.x >> 5);
    const int nWaves = gridDim.x * (blockDim.x >> 5);
    for (int e = wave; e < E; e += nWaves) {
        const int s = src[e], d = dst[e];
        const float* hp = hs + (size_t)s * 64 + 2 * lane;
        float* op = agg1 + (size_t)d * 64 + 2 * lane;
        atomicAdd(op + 0, hp[0]);
        atomicAdd(op + 1, hp[1]);
    }
}

// ---------------------------------------------------------------- conv1 GEMM via WMMA f32 16x16x4
// h1 = relu((agg1 * inv_in) @ W1 + b1).
// W1 staged TRANSPOSED in LDS: Wt[n][k] = W1[k][n] -> B fragment is one contiguous v2f.
// A layout (16x4 f32): lane L holds row M=L%16, K = 2*(L/16)+{0,1} per K-step.
// C/D layout: lane L (hi=L/16, n=L%16) holds D[i+8*hi][n] in vgpr i.
// ONE TILE PER WAVE: no outer loop, so LDS B-fragments cannot be hoisted+spilled.
__global__ void conv1_wmma_kernel(const float* __restrict__ agg, const float* __restrict__ inv_in,
                                  const float* __restrict__ W, const float* __restrict__ bias,
                                  float* __restrict__ h1, int nTiles, int N) {
    __shared__ float Wt[64 * 64];  // transposed W1
    for (int i = threadIdx.x; i < 64 * 64; i += blockDim.x) {
        const int k = i >> 6, n = i & 63;
        Wt[n * 64 + k] = W[i];
    }
    __syncthreads();
    const int lane = threadIdx.x & 31;
    const int hi = lane >> 4;
    const int ln = lane & 15;
    const int t = blockIdx.x * (blockDim.x >> 5) + (threadIdx.x >> 5);
    if (t < nTiles) {
        int arow = t * 16 + ln;
        if (arow >= N) arow = N - 1;
        const float sc = inv_in[arow];
        const float* ap = agg + (size_t)arow * 64;
        v8f acc[4] = {};
#pragma unroll
        for (int kk = 0; kk < 16; ++kk) {
            const int k = kk * 4 + 2 * hi;
            v2f a = *(const v2f*)(ap + k);
            a.x *= sc;
            a.y *= sc;
#pragma unroll
            for (int nt = 0; nt < 4; ++nt) {
                const v2f b = *(const v2f*)(&Wt[(nt * 16 + ln) * 64 + k]);
                acc[nt] = __builtin_amdgcn_wmma_f32_16x16x4_f32(
                    false, a, false, b, (short)0, acc[nt], false, false);
            }
        }
#pragma unroll
        for (int i = 0; i < 8; ++i) {
            const int m = t * 16 + i + 8 * hi;
            if (m < N) {
                float* orow = h1 + (size_t)m * 64;
#pragma unroll
                for (int nt = 0; nt < 4; ++nt) {
                    float v = acc[nt][i] + bias[nt * 16 + ln];
                    orow[nt * 16 + ln] = v > 0.0f ? v : 0.0f;
                }
            }
        }
    }
}

// ---------------------------------------------------------------- fused GRU + fc via WMMA
// gx = h1 @ w_ih.T (B[k][g] = w_ih[g*64+k]: contiguous in k -> v2f LDS loads)
// gates: r=sig(gx_r+bih_r+bhh_r), z=sig(gx_z+bih_z+bhh_z), n=tanh(gx_n+bih_n+r*bhh_n)
// out[m] = sum_j (1-z)*n * fc_w[j] + fc_b  (shuffle reduction over 16 lanes)
// ONE TILE PER WAVE (same anti-spill reasoning as conv1).
__global__ void gru_wmma_kernel(const float* __restrict__ h1, const float* __restrict__ w_ih,
                                const float* __restrict__ b_ih, const float* __restrict__ b_hh,
                                const float* __restrict__ fcw, const float* __restrict__ fcb,
                                float* __restrict__ out, int nTiles, int N) {
    __shared__ float Wl[192 * 64];  // 48 KB, w_ih row-major: Wl[g*64 + k]
    for (int i = threadIdx.x; i < 192 * 64; i += blockDim.x) Wl[i] = w_ih[i];
    __syncthreads();
    const int lane = threadIdx.x & 31;
    const int hi = lane >> 4;
    const int ln = lane & 15;
    const int t = blockIdx.x * (blockDim.x >> 5) + (threadIdx.x >> 5);
    if (t < nTiles) {
        int arow = t * 16 + ln;
        if (arow >= N) arow = N - 1;
        const float* ap = h1 + (size_t)arow * 64;
        v8f acc[12] = {};
#pragma unroll
        for (int kk = 0; kk < 16; ++kk) {
            const int k = kk * 4 + 2 * hi;
            const v2f a = *(const v2f*)(ap + k);
#pragma unroll
            for (int gt = 0; gt < 12; ++gt) {
                const v2f b = *(const v2f*)(&Wl[(gt * 16 + ln) * 64 + k]);
                acc[gt] = __builtin_amdgcn_wmma_f32_16x16x4_f32(
                    false, a, false, b, (short)0, acc[gt], false, false);
            }
        }
        const float fb = fcb[0];
#pragma unroll
        for (int i = 0; i < 8; ++i) {
            float partial = 0.0f;
#pragma unroll
            for (int nt = 0; nt < 4; ++nt) {
                const int j = nt * 16 + ln;
                const float gr = acc[nt][i] + b_ih[j] + b_hh[j];
                const float gz = acc[nt + 4][i] + b_ih[64 + j] + b_hh[64 + j];
                const float gn = acc[nt + 8][i] + b_ih[128 + j];
                const float r = 1.0f / (1.0f + __expf(-gr));
                const float z = 1.0f / (1.0f + __expf(-gz));
                const float nn = tanhf(gn + r * b_hh[128 + j]);
                partial += (1.0f - z) * nn * fcw[j];
            }
            // reduce over the 16 lanes of this half-wave (bit 4 untouched)
            partial += __shfl_xor(partial, 1, 32);
            partial += __shfl_xor(partial, 2, 32);
            partial += __shfl_xor(partial, 4, 32);
            partial += __shfl_xor(partial, 8, 32);
            const int m = t * 16 + i + 8 * hi;
            if (ln == 0 && m < N) out[m] = partial + fb;
        }
    }
}

// ---------------------------------------------------------------- launcher
extern "C" void kernel_launch(void* const* d_in, const int* in_sizes, int n_in,
                              void* d_out, int out_size, void* d_ws, size_t ws_size,
                              hipStream_t stream) {
    const float* feat  = (const float*)d_in[0];
    const int*   esrc  = (const int*)d_in[1];
    const int*   edst  = (const int*)d_in[2];
    const float* W0    = (const float*)d_in[3];
    const float* b0    = (const float*)d_in[4];
    const float* W1    = (const float*)d_in[5];
    const float* b1    = (const float*)d_in[6];
    const float* w_ih  = (const float*)d_in[7];
    // d_in[8] = w_hh: unused (h0 == 0 in the reference)
    const float* b_ih  = (const float*)d_in[9];
    const float* b_hh  = (const float*)d_in[10];
    const float* fcw   = (const float*)d_in[11];
    const float* fcb   = (const float*)d_in[12];
    float* out = (float*)d_out;

    const int N = in_sizes[0] / 3;
    const int E = in_sizes[1];

    // workspace layout (floats): deg_out N | deg_in N | agg0 4N | agg1 64N | hs/h1 64N
    const size_t needed = (size_t)134 * (size_t)N * sizeof(float);
    if (ws_size < needed) return;  // cannot allocate inside graph capture
    float* ws      = (float*)d_ws;
    float* deg_out = ws;                           // becomes inv_out after invsqrt
    float* deg_in  = ws + (size_t)N;               // becomes inv_in
    float* agg0    = ws + (size_t)2 * N;           // 4N (stride-4 padded)
    float* agg1    = agg0 + (size_t)4 * N;         // 64N
    float* hs      = agg1 + (size_t)64 * N;        // 64N; reused as h1

    const int nTiles = (N + 15) / 16;
    const int wmmaBlocks = (nTiles + 7) / 8;       // 8 waves per 256-thread block, 1 tile/wave

    // 1) zero accumulation buffers (deg_out, deg_in, agg0, agg1 = 70N contiguous floats)
    zero_f32_kernel<<<4096, 256, 0, stream>>>(ws, (long long)70 * N);
    // 2) degrees
    degree_kernel<<<2048, 256, 0, stream>>>(esrc, edst, deg_out, deg_in, E);
    // 3) inv-sqrt of clipped degrees, in place over both arrays
    invsqrt_kernel<<<784, 256, 0, stream>>>(ws, 2 * N);
    // 4) conv0 aggregation at F=3 (aggregate before matmul: A(hW) == (Ah)W)
    agg3_kernel<<<2048, 256, 0, stream>>>(feat, esrc, edst, deg_out, agg0, E);
    // 5) conv0 matmul [N,3]@[3,64] + relu, pre-scaled by inv_out for conv1
    conv0_kernel<<<8192, 256, 0, stream>>>(agg0, deg_in, deg_out, W0, b0, hs, N);
    // 6) conv1 aggregation at F=64: wave/edge, lane/2-floats, L2-resident atomics
    agg64_kernel<<<16384, 256, 0, stream>>>(hs, esrc, edst, agg1, E);
    // 7) conv1 GEMM via v_wmma_f32_16x16x4_f32 (fused inv_in scale + bias + relu)
    conv1_wmma_kernel<<<wmmaBlocks, 256, 0, stream>>>(agg1, deg_in, W1, b1, hs, nTiles, N);
    // 8) fused GRU GEMM (12 WMMA N-tiles) + gates + fc projection
    gru_wmma_kernel<<<wmmaBlocks, 256, 0, stream>>>(hs, w_ih, b_ih, b_hh, fcw, fcb, out, nTiles, N);
}